// RevolutionNaive1_30932354466108
// MI455X (gfx1250) — compile-verified
//
#include <hip/hip_runtime.h>
#include <math.h>

#define HH 56
#define WWI 56
#define HWP (HH*WWI)      // 3136
#define CC 256
#define GG 16
#define GCC 16
#define OUT1 96
#define KIN 2112
#define KT4 (KIN/4)       // 528
#define BB 4

typedef __attribute__((ext_vector_type(2))) float v2f;
typedef __attribute__((ext_vector_type(4))) float v4f;
typedef __attribute__((ext_vector_type(8))) float v8f;

// Pooled descriptor feature kidx (0..2111) at pixel (h,w) of batch b.
// Layout matches reference concat([x1(576), x2(768), x3(768)]).
__device__ __forceinline__ float pooled_feat(const float* __restrict__ x,
                                             int b, int kidx, int h, int w) {
    float v = 0.0f;
    if (kidx < 576) {
        // x1: max over channel sub-groups s=0..3 at a single tap
        int g  = kidx / 36;
        int r  = kidx - g * 36;
        int c4 = r / 9;
        int rr = r - c4 * 9;
        int kh = rr / 3;
        int kw = rr - kh * 3;
        int hh = h + kh - 1, ww = w + kw - 1;
        if ((unsigned)hh < (unsigned)HH && (unsigned)ww < (unsigned)WWI) {
            const float* p = x + (((size_t)b * CC + g * GCC + c4) * HH + hh) * WWI + ww;
            float a0 = p[0];
            float a1 = p[4 * HWP];
            float a2 = p[8 * HWP];
            float a3 = p[12 * HWP];
            v = fmaxf(fmaxf(a0, a1), fmaxf(a2, a3));
        }
    } else if (kidx < 1344) {
        // x2: max over kh (padded zeros participate)
        int t = kidx - 576;
        int g = t / 48;
        int r = t - g * 48;
        int c = r / 3;
        int kw = r - c * 3;
        int ww = w + kw - 1;
        if ((unsigned)ww < (unsigned)WWI) {
            const float* p = x + (((size_t)b * CC + g * GCC + c) * HH) * WWI + ww;
            int h0 = h - 1, h2 = h + 1;
            float t0 = ((unsigned)h0 < (unsigned)HH) ? p[h0 * WWI] : 0.0f;
            float t1 = p[h * WWI];
            float t2 = ((unsigned)h2 < (unsigned)HH) ? p[h2 * WWI] : 0.0f;
            v = fmaxf(fmaxf(t0, t1), t2);
        }
    } else {
        // x3: max over kw
        int t = kidx - 1344;
        int g = t / 48;
        int r = t - g * 48;
        int c = r / 3;
        int kh = r - c * 3;
        int hh = h + kh - 1;
        if ((unsigned)hh < (unsigned)HH) {
            const float* p = x + (((size_t)b * CC + g * GCC + c) * HH + hh) * WWI + w;
            float t0 = (w - 1 >= 0) ? p[-1] : 0.0f;
            float t1 = p[0];
            float t2 = (w + 1 < WWI) ? p[1] : 0.0f;
            v = fmaxf(fmaxf(t0, t1), t2);
        }
    }
    return v;
}

// Stage the pooled tensor into quad-of-K layout: P4[bl][kt][pixel][4].
// One thread computes 4 consecutive K features and stores one b128.
__global__ __launch_bounds__(256) void k_pool(
    const float* __restrict__ x, float* __restrict__ P4, int b0, int nb) {
    int idx = blockIdx.x * 256 + threadIdx.x;
    int total = nb * KT4 * HWP;
    if (idx >= total) return;
    int bl = idx / (KT4 * HWP);
    int r  = idx - bl * (KT4 * HWP);
    int kt = r / HWP;
    int p  = r - kt * HWP;
    int h = p / WWI, w = p - h * WWI;
    int b = b0 + bl;
    v4f v;
    v[0] = pooled_feat(x, b, kt * 4 + 0, h, w);
    v[1] = pooled_feat(x, b, kt * 4 + 1, h, w);
    v[2] = pooled_feat(x, b, kt * 4 + 2, h, w);
    v[3] = pooled_feat(x, b, kt * 4 + 3, h, w);
    *(v4f*)(P4 + (size_t)idx * 4) = v;
}

// conv1 (96x2112 GEMM) from staged P4, + BN + ReLU.
// One wave -> 16 pixels x 48 output channels (3 WMMA M-tiles).
// Inner loop: 1 b64 B-load + 3 b64 A-loads + 3 wmma.
__global__ __launch_bounds__(256) void k_gemm1s(
    const float* __restrict__ P4, const float* __restrict__ w1,
    const float* __restrict__ gamma, const float* __restrict__ beta,
    const float* __restrict__ mean, const float* __restrict__ var,
    float* __restrict__ out1, int nunits) {
    const int lane = threadIdx.x & 31;
    const int wid  = threadIdx.x >> 5;
    int unit = blockIdx.x * 8 + wid;
    if (unit >= nunits) return;           // wave-uniform
    int bl    = unit / 392;
    int rem   = unit - bl * 392;
    int ntile = rem >> 1;
    int mgrp  = rem & 1;
    int m_base = mgrp * 48;

    const int li = lane & 15;
    const int hf = lane >> 4;
    const int p  = ntile * 16 + li;

    v8f acc[3] = {};
    const float* w1r = w1 + (size_t)(m_base + li) * KIN;
    const float* Pb  = P4 + ((size_t)bl * KT4 * HWP + p) * 4 + hf * 2;

    for (int kt = 0; kt < KT4; ++kt) {
        v2f bv = *(const v2f*)(Pb + (size_t)kt * HWP * 4);
        int k = kt * 4 + hf * 2;          // lanes 0-15: K=0,1  lanes 16-31: K=2,3
        v2f a0 = *(const v2f*)(w1r + 0 * 16 * KIN + k);
        v2f a1 = *(const v2f*)(w1r + 1 * 16 * KIN + k);
        v2f a2 = *(const v2f*)(w1r + 2 * 16 * KIN + k);
        acc[0] = __builtin_amdgcn_wmma_f32_16x16x4_f32(false, a0, false, bv, (short)0, acc[0], false, false);
        acc[1] = __builtin_amdgcn_wmma_f32_16x16x4_f32(false, a1, false, bv, (short)0, acc[1], false, false);
        acc[2] = __builtin_amdgcn_wmma_f32_16x16x4_f32(false, a2, false, bv, (short)0, acc[2], false, false);
    }

    // C layout: vgpr r, lanes 0-15 -> M=r, lanes 16-31 -> M=r+8
    #pragma unroll
    for (int mt = 0; mt < 3; ++mt) {
        #pragma unroll
        for (int r = 0; r < 8; ++r) {
            int m = m_base + mt * 16 + r + hf * 8;
            float iv = gamma[m] * rsqrtf(var[m] + 1e-5f);
            float sh = beta[m] - mean[m] * iv;
            float vv = acc[mt][r] * iv + sh;
            out1[((size_t)bl * OUT1 + m) * HWP + p] = fmaxf(vv, 0.0f);
        }
    }
}

// Fallback: conv1 with pooling fused into B-operand (no P4 workspace needed).
__global__ __launch_bounds__(256) void k_gemm1_fused(
    const float* __restrict__ x, const float* __restrict__ w1,
    const float* __restrict__ gamma, const float* __restrict__ beta,
    const float* __restrict__ mean, const float* __restrict__ var,
    float* __restrict__ out1) {
    const int lane = threadIdx.x & 31;
    const int wid  = threadIdx.x >> 5;
    int unit = blockIdx.x * 8 + wid;
    if (unit >= BB * 196 * 2) return;
    int b     = unit / 392;
    int rem   = unit - b * 392;
    int ntile = rem >> 1;
    int mgrp  = rem & 1;
    int m_base = mgrp * 48;

    const int li = lane & 15;
    const int hf = lane >> 4;
    const int p  = ntile * 16 + li;
    const int h  = p / WWI, w = p - h * WWI;

    v8f acc[3] = {};
    const float* w1r = w1 + (size_t)(m_base + li) * KIN;

    for (int kt = 0; kt < KT4; ++kt) {
        int k = kt * 4 + hf * 2;
        v2f bv;
        bv[0] = pooled_feat(x, b, k,     h, w);
        bv[1] = pooled_feat(x, b, k + 1, h, w);
        v2f a0 = *(const v2f*)(w1r + 0 * 16 * KIN + k);
        v2f a1 = *(const v2f*)(w1r + 1 * 16 * KIN + k);
        v2f a2 = *(const v2f*)(w1r + 2 * 16 * KIN + k);
        acc[0] = __builtin_amdgcn_wmma_f32_16x16x4_f32(false, a0, false, bv, (short)0, acc[0], false, false);
        acc[1] = __builtin_amdgcn_wmma_f32_16x16x4_f32(false, a1, false, bv, (short)0, acc[1], false, false);
        acc[2] = __builtin_amdgcn_wmma_f32_16x16x4_f32(false, a2, false, bv, (short)0, acc[2], false, false);
    }

    #pragma unroll
    for (int mt = 0; mt < 3; ++mt) {
        #pragma unroll
        for (int r = 0; r < 8; ++r) {
            int m = m_base + mt * 16 + r + hf * 8;
            float iv = gamma[m] * rsqrtf(var[m] + 1e-5f);
            float sh = beta[m] - mean[m] * iv;
            float vv = acc[mt][r] * iv + sh;
            out1[((size_t)b * OUT1 + m) * HWP + p] = fmaxf(vv, 0.0f);
        }
    }
}

// conv2 and conv3 (96x96 GEMMs) sharing the B-operand, + bias.
__global__ __launch_bounds__(256) void k_gemm23(
    const float* __restrict__ out1,
    const float* __restrict__ w2, const float* __restrict__ b2,
    const float* __restrict__ w3, const float* __restrict__ b3,
    float* __restrict__ wa, float* __restrict__ wb) {
    const int lane = threadIdx.x & 31;
    const int wid  = threadIdx.x >> 5;
    int unit = blockIdx.x * 8 + wid;
    if (unit >= BB * 196 * 2) return;
    int b     = unit / 392;
    int rem   = unit - b * 392;
    int ntile = rem >> 1;
    int mgrp  = rem & 1;
    int m_base = mgrp * 48;

    const int li = lane & 15;
    const int hf = lane >> 4;
    const int p  = ntile * 16 + li;

    v8f accA[3] = {};
    v8f accB[3] = {};
    const float* brow = out1 + (size_t)b * OUT1 * HWP + p;

    for (int kt = 0; kt < OUT1 / 4; ++kt) {
        int k = kt * 4 + hf * 2;
        v2f bv;
        bv[0] = brow[(size_t)k * HWP];
        bv[1] = brow[(size_t)(k + 1) * HWP];
        #pragma unroll
        for (int mt = 0; mt < 3; ++mt) {
            int row = m_base + mt * 16 + li;
            v2f aA = *(const v2f*)(w2 + (size_t)row * OUT1 + k);
            v2f aB = *(const v2f*)(w3 + (size_t)row * OUT1 + k);
            accA[mt] = __builtin_amdgcn_wmma_f32_16x16x4_f32(false, aA, false, bv, (short)0, accA[mt], false, false);
            accB[mt] = __builtin_amdgcn_wmma_f32_16x16x4_f32(false, aB, false, bv, (short)0, accB[mt], false, false);
        }
    }

    #pragma unroll
    for (int mt = 0; mt < 3; ++mt) {
        #pragma unroll
        for (int r = 0; r < 8; ++r) {
            int m = m_base + mt * 16 + r + hf * 8;
            wa[((size_t)b * OUT1 + m) * HWP + p] = accA[mt][r] + b2[m];
            wb[((size_t)b * OUT1 + m) * HWP + p] = accB[mt][r] + b3[m];
        }
    }
}

// Dynamic separable kernel: outer product, softmax over 9 taps, weighted mean,
// 2x2 pixel-shuffle upsample.
__global__ __launch_bounds__(256) void k_apply(
    const float* __restrict__ x, const float* __restrict__ wa,
    const float* __restrict__ wb, float* __restrict__ out) {
    int idx = blockIdx.x * blockDim.x + threadIdx.x;
    if (idx >= BB * GG * HWP) return;
    int b = idx / (GG * HWP);
    int r = idx - b * GG * HWP;
    int g = r / HWP;
    int p = r - g * HWP;
    int h = p / WWI, w = p - h * WWI;

    float av[3][2], bvv[3][2];
    const float* wap = wa + ((size_t)b * OUT1 + g * 6) * HWP + p;
    const float* wbp = wb + ((size_t)b * OUT1 + g * 6) * HWP + p;
    #pragma unroll
    for (int kh = 0; kh < 3; ++kh)
        #pragma unroll
        for (int i = 0; i < 2; ++i) {
            av[kh][i]  = wap[(kh * 2 + i) * HWP];
            bvv[kh][i] = wbp[(kh * 2 + i) * HWP];
        }

    float sm[9][2][2];
    #pragma unroll
    for (int i = 0; i < 2; ++i)
        #pragma unroll
        for (int j = 0; j < 2; ++j) {
            float t[9];
            float mx = -1e30f;
            #pragma unroll
            for (int kh = 0; kh < 3; ++kh)
                #pragma unroll
                for (int kw = 0; kw < 3; ++kw) {
                    float vv = av[kh][i] * bvv[kw][j];
                    t[kh * 3 + kw] = vv;
                    mx = fmaxf(mx, vv);
                }
            float s = 0.0f;
            #pragma unroll
            for (int kk = 0; kk < 9; ++kk) { float e = __expf(t[kk] - mx); t[kk] = e; s += e; }
            float rs = 1.0f / s;
            #pragma unroll
            for (int kk = 0; kk < 9; ++kk) sm[kk][i][j] = t[kk] * rs;
        }

    int off[9];
    bool valid[9];
    #pragma unroll
    for (int kh = 0; kh < 3; ++kh)
        #pragma unroll
        for (int kw = 0; kw < 3; ++kw) {
            int hh = h + kh - 1, ww = w + kw - 1;
            valid[kh * 3 + kw] = ((unsigned)hh < (unsigned)HH) && ((unsigned)ww < (unsigned)WWI);
            off[kh * 3 + kw] = hh * WWI + ww;
        }

    const float ninth = 1.0f / 9.0f;
    for (int c = 0; c < GCC; ++c) {
        const float* xb = x + ((size_t)b * CC + g * GCC + c) * HWP;
        float a00 = 0.f, a01 = 0.f, a10 = 0.f, a11 = 0.f;
        #pragma unroll
        for (int kk = 0; kk < 9; ++kk) {
            float xv = valid[kk] ? xb[off[kk]] : 0.0f;
            a00 += xv * sm[kk][0][0];
            a01 += xv * sm[kk][0][1];
            a10 += xv * sm[kk][1][0];
            a11 += xv * sm[kk][1][1];
        }
        size_t ob = ((size_t)b * CC + g * GCC + c);
        out[(ob * 112 + (2 * h + 0)) * 112 + (2 * w + 0)] = a00 * ninth;
        out[(ob * 112 + (2 * h + 0)) * 112 + (2 * w + 1)] = a01 * ninth;
        out[(ob * 112 + (2 * h + 1)) * 112 + (2 * w + 0)] = a10 * ninth;
        out[(ob * 112 + (2 * h + 1)) * 112 + (2 * w + 1)] = a11 * ninth;
    }
}

extern "C" void kernel_launch(void* const* d_in, const int* in_sizes, int n_in,
                              void* d_out, int out_size, void* d_ws, size_t ws_size,
                              hipStream_t stream) {
    const float* x     = (const float*)d_in[0];
    const float* w1    = (const float*)d_in[1];
    const float* gamma = (const float*)d_in[2];
    const float* beta  = (const float*)d_in[3];
    const float* mean  = (const float*)d_in[4];
    const float* var   = (const float*)d_in[5];
    const float* w2    = (const float*)d_in[6];
    const float* b2    = (const float*)d_in[7];
    const float* w3    = (const float*)d_in[8];
    const float* b3    = (const float*)d_in[9];
    float* out = (float*)d_out;

    const size_t per      = (size_t)BB * OUT1 * HWP;        // 1,204,224 floats
    const size_t small_b  = 3 * per * sizeof(float);        // out1 + wa + wb
    const size_t p_one    = (size_t)KT4 * HWP * 4;          // floats per batch
    const size_t p_full_b = (size_t)BB * p_one * sizeof(float);
    const size_t p_one_b  = p_one * sizeof(float);

    float* out1 = (float*)d_ws;
    float* wa   = out1 + per;
    float* wb   = wa + per;
    float* P4   = wb + per;

    if (ws_size >= small_b + p_full_b) {
        // Full staged: pool all batches (P4 ~106MB, fits the 192MB L2), then GEMM.
        int pt = BB * KT4 * HWP;
        k_pool<<<(pt + 255) / 256, 256, 0, stream>>>(x, P4, 0, BB);
        k_gemm1s<<<196, 256, 0, stream>>>(P4, w1, gamma, beta, mean, var, out1, BB * 392);
    } else if (ws_size >= small_b + p_one_b) {
        // Per-batch staging, reusing one P4 region (stream-ordered).
        int pt = KT4 * HWP;
        for (int b = 0; b < BB; ++b) {
            k_pool<<<(pt + 255) / 256, 256, 0, stream>>>(x, P4, b, 1);
            k_gemm1s<<<49, 256, 0, stream>>>(P4, w1, gamma, beta, mean, var,
                                             out1 + (size_t)b * OUT1 * HWP, 392);
        }
    } else {
        // Minimal-workspace fallback: pooling fused into the GEMM B-operand.
        k_gemm1_fused<<<196, 256, 0, stream>>>(x, w1, gamma, beta, mean, var, out1);
    }

    k_gemm23<<<196, 256, 0, stream>>>(out1, w2, b2, w3, b3, wa, wb);

    int total = BB * GG * HWP;                              // 200,704 threads
    k_apply<<<(total + 255) / 256, 256, 0, stream>>>(x, wa, wb, out);
}